// HeteroFASTGCN_86861418594479
// MI455X (gfx1250) — compile-verified
//
#include <hip/hip_runtime.h>
#include <hip/hip_bf16.h>
#include <math.h>

#define NODE_NUM 8192
#define SEQ_LEN  192
#define NN       (NODE_NUM + SEQ_LEN)   // 8384 nodes
#define C        128
#define B        2
#define E        262144

typedef __attribute__((ext_vector_type(2))) float v2f;
typedef __attribute__((ext_vector_type(8))) float v8f;

// ---------------------------------------------------------------------------
// Degree counting: every edge is exactly one of {ss,tt} (intra) or {st,ts}
// (inter), decided by region equality. Degrees depend only on edge_index, so
// this runs once and feeds all 4 conv steps.
// ---------------------------------------------------------------------------
__global__ __launch_bounds__(256) void deg_kernel(const int* __restrict__ ei,
                                                  unsigned* __restrict__ deg_intra,
                                                  unsigned* __restrict__ deg_inter) {
    int idx = blockIdx.x * blockDim.x + threadIdx.x;       // [0, B*E)
    int b = idx / E, e = idx - b * E;
    const int* eb = ei + (size_t)b * 2 * E;
    int src = eb[e];
    int dst = eb[E + e];
    bool ss = src < NODE_NUM, sd = dst < NODE_NUM;
    if (ss == sd) atomicAdd(&deg_intra[b * NN + dst], 1u);
    else          atomicAdd(&deg_inter[b * NN + dst], 1u);
}

__global__ __launch_bounds__(256) void norm_kernel(const unsigned* __restrict__ degI,
                                                   const unsigned* __restrict__ degE,
                                                   float* __restrict__ dinvI,
                                                   float* __restrict__ invdegI,
                                                   float* __restrict__ dinvE) {
    int idx = blockIdx.x * blockDim.x + threadIdx.x;
    if (idx >= B * NN) return;
    float d = (float)(degI[idx] + 1u);      // gcn_norm with self loop: deg+1
    dinvI[idx]   = 1.0f / sqrtf(d);
    invdegI[idx] = 1.0f / d;
    unsigned de = degE[idx];
    dinvE[idx] = de ? (1.0f / (float)de) : 0.0f;  // directed in-degree norm
}

// ---------------------------------------------------------------------------
// Gate GEMM via fp32 WMMA: G[n][0..3] = H[n,:] . {WA[:C], WA[C:], WB[:C], WB[C:]}
// One wave per 16-node tile; 32 accumulating v_wmma_f32_16x16x4_f32 cover K=128.
// A layout (16x4 f32): lanes 0-15 hold K=0(v0),K=1(v1); lanes 16-31 K=2,K=3.
// B layout (4x16 f32): v0 = row K=0 (lanes 0-15) / K=2 (lanes 16-31); v1 = K=1/K=3.
// D layout (16x16 f32): lane = {half:N}, VGPR v = row M=v+8*half.
// Lanes with col>=4 replicate column (col&3): D columns are independent, and
// we only store cols 0..3 -- this keeps EXEC uniform (no per-load exec masking).
// ---------------------------------------------------------------------------
__global__ __launch_bounds__(256) void gates_wmma_kernel(const float* __restrict__ h,
                                                         const float* __restrict__ WA,
                                                         const float* __restrict__ WB,
                                                         float* __restrict__ G) {
    int wave = (int)((blockIdx.x * blockDim.x + threadIdx.x) >> 5);  // tile id
    int lane = threadIdx.x & 31;
    int base = wave * 16;                 // flat node base, B*NN divisible by 16
    int m    = lane & 15;
    int kh   = lane >> 4;                 // 0 or 1 (K sub-pair select)
    const float* row = h + (size_t)(base + m) * C;

    int col = lane & 15;                  // B column; only 0..3 are real
    // col&2 selects W-matrix, col&1 selects [:C] vs [C:]; branch-free.
    const float* wcol = ((col & 2) ? WB : WA) + ((col & 1) ? C : 0);

    v8f acc = {};
    for (int k = 0; k < C; k += 4) {
        v2f a, bm;
        a.x  = row[k + 2 * kh];
        a.y  = row[k + 2 * kh + 1];
        bm.x = wcol[k + 2 * kh];
        bm.y = wcol[k + 2 * kh + 1];
        acc = __builtin_amdgcn_wmma_f32_16x16x4_f32(
            /*neg_a=*/false, a, /*neg_b=*/false, bm,
            /*c_mod=*/(short)0, acc, /*reuse_a=*/false, /*reuse_b=*/false);
    }
    if (col < 4) {
#pragma unroll
        for (int v = 0; v < 8; ++v) {
            int rowm = v + 8 * kh;
            G[(size_t)(base + rowm) * 4 + col] = acc[v];
        }
    }
}

// Region-based selection of gi/gj + self-loop gate (intra only).
__global__ __launch_bounds__(256) void finalize_gates_kernel(const float* __restrict__ G,
                                                             const float* __restrict__ invdegI,
                                                             float* __restrict__ gi,
                                                             float* __restrict__ gj,
                                                             float* __restrict__ aself,
                                                             int intra) {
    int idx = blockIdx.x * blockDim.x + threadIdx.x;
    if (idx >= B * NN) return;
    int n = idx % NN;
    bool s = n < NODE_NUM;
    float d0 = G[(size_t)idx * 4 + 0];
    float d1 = G[(size_t)idx * 4 + 1];
    float d2 = G[(size_t)idx * 4 + 2];
    float d3 = G[(size_t)idx * 4 + 3];
    float vgj = s ? d1 : d3;              // src-role gate (both intra & inter)
    float vgi;
    if (intra) {
        vgi = s ? d0 : d2;                // dst s -> W_ss, dst t -> W_tt
        aself[idx] = tanhf(vgi + vgj) * invdegI[idx];
    } else {
        vgi = s ? d2 : d0;                // dst s -> W_ts, dst t -> W_st
    }
    gi[idx] = vgi;
    gj[idx] = vgj;
}

// ---------------------------------------------------------------------------
// Edge scatter: one wave32 per edge; 4 floats/lane; hardware f32 atomics into
// acc (L2-resident, ~8.6MB per batch). Edge decode is wave-uniform.
// ---------------------------------------------------------------------------
__global__ __launch_bounds__(256) void scatter_kernel(const float* __restrict__ h,
                                                      const int* __restrict__ ei,
                                                      const float* __restrict__ gi,
                                                      const float* __restrict__ gj,
                                                      const float* __restrict__ dinvI,
                                                      const float* __restrict__ dinvE,
                                                      float* __restrict__ acc,
                                                      int intra) {
    int wave = (int)((blockIdx.x * blockDim.x + threadIdx.x) >> 5);  // [0, B*E)
    int lane = threadIdx.x & 31;
    int b = wave / E, e = wave - b * E;
    const int* eb = ei + (size_t)b * 2 * E;
    int src = eb[e];
    int dst = eb[E + e];
    bool ss = src < NODE_NUM, sd = dst < NODE_NUM;
    bool valid = intra ? (ss == sd) : (ss != sd);
    if (!valid) return;                    // wave-uniform exit

    int ns = b * NN + src;
    int nd = b * NN + dst;
    const float4* xr = (const float4*)(h + (size_t)ns * C);
    __builtin_prefetch(&xr[lane], 0, 1);   // global_prefetch_b8 on source row

    float a    = tanhf(gi[nd] + gj[ns]);
    float coef = intra ? (a * dinvI[ns] * dinvI[nd]) : (a * dinvE[nd]);

    float4 xv = xr[lane];
    float* o = acc + (size_t)nd * C + lane * 4;
    __hip_atomic_fetch_add(&o[0], xv.x * coef, __ATOMIC_RELAXED, __HIP_MEMORY_SCOPE_AGENT);
    __hip_atomic_fetch_add(&o[1], xv.y * coef, __ATOMIC_RELAXED, __HIP_MEMORY_SCOPE_AGENT);
    __hip_atomic_fetch_add(&o[2], xv.z * coef, __ATOMIC_RELAXED, __HIP_MEMORY_SCOPE_AGENT);
    __hip_atomic_fetch_add(&o[3], xv.w * coef, __ATOMIC_RELAXED, __HIP_MEMORY_SCOPE_AGENT);
}

// Residual + (intra) self-loop + optional relu; float4 vectorized (b128 vmem).
__global__ __launch_bounds__(256) void combine_kernel(const float* __restrict__ h,
                                                      const float* __restrict__ acc,
                                                      const float* __restrict__ aself,
                                                      float* __restrict__ out,
                                                      int intra, int relu) {
    size_t q = (size_t)blockIdx.x * blockDim.x + threadIdx.x;   // float4 index
    if (q >= (size_t)B * NN * (C / 4)) return;
    size_t node = q / (C / 4);
    const float4* h4 = (const float4*)h;
    const float4* a4 = (const float4*)acc;
    float4* o4 = (float4*)out;
    float4 hv = h4[q];
    float4 av = a4[q];
    float sg = intra ? aself[node] : 0.0f;
    float4 v;
    v.x = hv.x + av.x + hv.x * sg;
    v.y = hv.y + av.y + hv.y * sg;
    v.z = hv.z + av.z + hv.z * sg;
    v.w = hv.w + av.w + hv.w * sg;
    if (relu) {
        v.x = fmaxf(v.x, 0.0f); v.y = fmaxf(v.y, 0.0f);
        v.z = fmaxf(v.z, 0.0f); v.w = fmaxf(v.w, 0.0f);
    }
    o4[q] = v;
}

extern "C" void kernel_launch(void* const* d_in, const int* in_sizes, int n_in,
                              void* d_out, int out_size, void* d_ws, size_t ws_size,
                              hipStream_t stream) {
    const float* x   = (const float*)d_in[0];   // [B, N, C]
    const int*   ei  = (const int*)d_in[1];     // [B, 2, E]
    const float* Wss = (const float*)d_in[2];   // [2, 256]
    const float* Wtt = (const float*)d_in[3];
    const float* Wst = (const float*)d_in[4];
    const float* Wts = (const float*)d_in[5];
    float* out = (float*)d_out;                 // [B, N, C]

    char* ws = (char*)d_ws;
    size_t off = 0;
    auto alloc = [&](size_t bytes) -> void* {
        off = (off + 255) & ~(size_t)255;
        void* p = ws + off;
        off += bytes;
        return p;
    };
    const size_t NODES = (size_t)B * NN;
    float*    hA      = (float*)alloc(NODES * C * sizeof(float));
    float*    accb    = (float*)alloc(NODES * C * sizeof(float));
    float*    G       = (float*)alloc(NODES * 4 * sizeof(float));
    float*    gi      = (float*)alloc(NODES * sizeof(float));
    float*    gj      = (float*)alloc(NODES * sizeof(float));
    float*    aself   = (float*)alloc(NODES * sizeof(float));
    float*    dinvI   = (float*)alloc(NODES * sizeof(float));
    float*    invdegI = (float*)alloc(NODES * sizeof(float));
    float*    dinvE   = (float*)alloc(NODES * sizeof(float));
    unsigned* degI    = (unsigned*)alloc(NODES * sizeof(unsigned));
    unsigned* degE    = (unsigned*)alloc(NODES * sizeof(unsigned));

    // Degrees: edge-structure only -> computed once, reused for all 4 steps.
    hipMemsetAsync(degI, 0, NODES * sizeof(unsigned), stream);
    hipMemsetAsync(degE, 0, NODES * sizeof(unsigned), stream);
    deg_kernel<<<(B * E) / 256, 256, 0, stream>>>(ei, degI, degE);
    norm_kernel<<<(int)((NODES + 255) / 256), 256, 0, stream>>>(degI, degE, dinvI, invdegI, dinvE);

    const float* hcur = x;
    float* hnext[4] = { hA, out, hA, out };     // final step lands in d_out

    for (int step = 0; step < 4; ++step) {
        int intra = (step % 2 == 0);
        int li    = step / 2;
        const float* WA = intra ? (Wss + li * 256) : (Wst + li * 256);
        const float* WB = intra ? (Wtt + li * 256) : (Wts + li * 256);

        // 1. gates (fp32 WMMA GEMM): 1048 waves, 131 blocks of 256 exactly
        gates_wmma_kernel<<<(int)((NODES / 16) * 32 / 256), 256, 0, stream>>>(hcur, WA, WB, G);
        finalize_gates_kernel<<<(int)((NODES + 255) / 256), 256, 0, stream>>>(
            G, invdegI, gi, gj, aself, intra);

        // 2. scatter
        hipMemsetAsync(accb, 0, NODES * C * sizeof(float), stream);
        scatter_kernel<<<(int)(((size_t)B * E * 32) / 256), 256, 0, stream>>>(
            hcur, ei, gi, gj, dinvI, dinvE, accb, intra);

        // 3. combine (residual + self loop + relu on all but last step)
        int relu = (step < 3);
        combine_kernel<<<(int)((NODES * (C / 4) + 255) / 256), 256, 0, stream>>>(
            hcur, accb, aself, hnext[step], intra, relu);

        hcur = hnext[step];
    }
}